// Transformer2D_36876589204149
// MI455X (gfx1250) — compile-verified
//
#include <hip/hip_runtime.h>

// Flow-warp, nearest-neighbor, border clamp.
// src:  [B=8, C=16, H=512, W=512] f32
// flow: [B=8, 2,    H=512, W=512] f32   (ch0 = row/y disp, ch1 = col/x disp)
// out:  [B=8, C=16, H=512, W=512] f32
//
// Memory-bound: ~285 MB traffic -> ~12 us floor @ 23.3 TB/s. No contraction
// dimension -> WMMA inapplicable; optimized for the gfx1250 memory path:
// wave32 blocks, all-scalar (SADDR) base addressing with 24-bit immediates,
// global_prefetch_b8 at the identity position, 64 gathers in flight per
// thread, non-temporal b128 stores.

typedef __attribute__((ext_vector_type(4))) float v4f;

namespace {
constexpr int kB  = 8;
constexpr int kC  = 16;
constexpr int kH  = 512;
constexpr int kW  = 512;
constexpr int kHW  = kH * kW;          // 2^18 pixels per (b, c) plane
constexpr int kCHW = kC * kHW;
constexpr int kPixPerThread = 4;       // one b128 NT store per channel
constexpr int kBlockThreads = 256;     // 8 wave32 waves
constexpr int kPixPerBlock  = kBlockThreads * kPixPerThread;   // 1024
constexpr int kChSplit = 8;            // c*kHW*4 fits signed-24-bit IOFFSET for c<8
}

__global__ __launch_bounds__(kBlockThreads, 8)
void warp_nearest_kernel(const float* __restrict__ src,
                         const float* __restrict__ flow,
                         float* __restrict__ out) {
  // kHW / kPixPerBlock == 256 -> each block lies entirely inside one batch,
  // so batch index / pixel base are wave-uniform (SGPR): every global access
  // lowers to SADDR + 32-bit voffset, no per-lane 64-bit address arithmetic.
  const int bid = (int)blockIdx.x;
  const int b   = bid >> 8;                         // uniform batch index
  const int pb  = (bid & 255) << 10;                // uniform pixel base
  const int p   = pb + ((int)threadIdx.x << 2);     // per-lane pixel (x4)

  const int y = p >> 9;                             // kW == 2^9
  const int x = p & (kW - 1);

  // Uniform bases; explicit high-half bases keep channels 8..15 within the
  // signed 24-bit instruction offset (c*1MB < 8MB) instead of forcing the
  // compiler into per-lane 64-bit adds.
  const float* __restrict__ srcb    = src + (size_t)b * kCHW;
  const float* __restrict__ srcb_hi = srcb + (size_t)kChSplit * kHW;
  const float* __restrict__ flwb    = flow + (size_t)b * (2 * kHW);
  float*       __restrict__ outb    = out + (size_t)b * kCHW;
  float*       __restrict__ outb_hi = outb + (size_t)kChSplit * kHW;

  // gfx1250 speculative prefetch (global_prefetch_b8): flow ~ N(0,1), so the
  // gather target is almost always within the cacheline(s) at the identity
  // position. Warm L0/L2 for all 16 channels while the dependent flow load +
  // round/clamp index math is still in flight.
#pragma unroll
  for (int c = 0; c < kChSplit; ++c) {
    __builtin_prefetch(srcb    + (size_t)c * kHW + p, /*rw=*/0, /*locality=*/3);
    __builtin_prefetch(srcb_hi + (size_t)c * kHW + p, /*rw=*/0, /*locality=*/3);
  }

  // Vectorized flow loads: p is a multiple of 4 -> 16-byte aligned.
  const v4f f0 = *reinterpret_cast<const v4f*>(flwb + p);          // y disp
  const v4f f1 = *reinterpret_cast<const v4f*>(flwb + kHW + p);    // x disp

  int lin[kPixPerThread];
#pragma unroll
  for (int i = 0; i < kPixPerThread; ++i) {
    // Replicate the reference fp32 op sequence exactly:
    //   n   = 2*((idx + f)/(S-1) - 0.5)
    //   pix = (n + 1) * 0.5 * (S - 1)
    const float fy = f0[i];
    const float fx = f1[i];
    const float ny = 2.0f * (((float)y       + fy) / (float)(kH - 1) - 0.5f);
    const float nx = 2.0f * (((float)(x + i) + fx) / (float)(kW - 1) - 0.5f);
    const float ypix = (ny + 1.0f) * 0.5f * (float)(kH - 1);
    const float xpix = (nx + 1.0f) * 0.5f * (float)(kW - 1);
    // jnp.round == round half-to-even == v_rndne_f32
    int yi = (int)rintf(ypix);
    int xi = (int)rintf(xpix);
    yi = yi < 0 ? 0 : (yi > kH - 1 ? kH - 1 : yi);   // border clamp
    xi = xi < 0 ? 0 : (xi > kW - 1 ? kW - 1 : xi);
    lin[i] = (yi << 9) + xi;
  }

  // 16 channels x 4 gathers fully unrolled: 64 independent b32 loads in
  // flight off scalar bases, then 16 non-temporal b128 stores (write-once
  // output bypasses L2 retention, keeping L2 hot with src for cross-wave
  // gather reuse — all 134 MB of src fit in the 192 MB L2).
#pragma unroll
  for (int c = 0; c < kC; ++c) {
    const float* __restrict__ sc =
        (c < kChSplit) ? (srcb + (size_t)c * kHW)
                       : (srcb_hi + (size_t)(c - kChSplit) * kHW);
    float* __restrict__ oc =
        (c < kChSplit) ? (outb + (size_t)c * kHW + p)
                       : (outb_hi + (size_t)(c - kChSplit) * kHW + p);
    v4f v;
    v.x = sc[lin[0]];
    v.y = sc[lin[1]];
    v.z = sc[lin[2]];
    v.w = sc[lin[3]];
    __builtin_nontemporal_store(v, reinterpret_cast<v4f*>(oc));
  }
}

extern "C" void kernel_launch(void* const* d_in, const int* in_sizes, int n_in,
                              void* d_out, int out_size, void* d_ws, size_t ws_size,
                              hipStream_t stream) {
  (void)in_sizes; (void)n_in; (void)out_size; (void)d_ws; (void)ws_size;
  const float* src  = (const float*)d_in[0];
  const float* flow = (const float*)d_in[1];
  float*       out  = (float*)d_out;

  const dim3 block(kBlockThreads);
  const dim3 grid((kB * kHW) / kPixPerBlock);   // 2048 blocks, exact tiling
  hipLaunchKernelGGL(warp_nearest_kernel, grid, block, 0, stream, src, flow, out);
}